// IMUAugmentation_6476810682521
// MI455X (gfx1250) — compile-verified
//
#include <hip/hip_runtime.h>
#include <math.h>

// ---------------------------------------------------------------------------
// IMU augmentation: OU bias walk (constant-coefficient linear recurrence) +
// elementwise combine. Blocked parallel scan:
//   - staging: GLOBAL_LOAD_ASYNC_TO_LDS_B64 (raw bytes, ASYNCcnt path)
//   - intra-chunk prefix: V_WMMA_F32_16X16X4_F32, D = (ds*P) x N_raw with
//     P lower-triangular powers of `a`
// ---------------------------------------------------------------------------

namespace {

constexpr int kB       = 32;
constexpr int kL       = 65536;          // T*K
constexpr int kCH      = 2048;           // chunk of t-steps per block
constexpr int kNCH     = kL / kCH;       // 32 chunks per batch
constexpr int kThreads = 256;            // 8 wave32 per block
constexpr int kSegs    = kCH / 16;       // 128 segments of 16 steps

#define A_COEF ((float)(1.0 - (1.0 / 100.0) * 0.001))   // a = 1 - dt/tau = 0.99999
#define DIFFS  ((float)0.004472135954999579)            // sqrt(2/tau)*sqrt(dt)

typedef __attribute__((ext_vector_type(2))) float v2f;
typedef __attribute__((ext_vector_type(8))) float v8f;
typedef __attribute__((address_space(3))) float lds_f;  // raw LDS addressing

__device__ __forceinline__ float std0_of(int c)  { return c < 3 ? 0.2f   : 0.015f; }

__device__ __forceinline__ float powi_a(int e) {  // a^e, small e
  float r = 1.0f;
  for (int i = 0; i < e; ++i) r *= A_COEF;
  return r;
}

// Async-stage RAW eps_walk increments into LDS as [kCH][6]:
//   nlds[tl*6+c] = eps_walk[(b*(L-1) + t0+tl-1)*6 + c]     (raw, unscaled)
//   nlds[0..5]   = eps0[c] / DIFFS  when t0==0  (so folded ds gives eps0*std0)
// The per-channel scale ds_c = std0*DIFFS is folded in by the consumers.
__device__ __forceinline__ void stage_n_async(float* nlds,
                                              const float* __restrict__ eps_walk,
                                              const float* __restrict__ eps0,
                                              int b, int t0, int k) {
  long long gstart = ((long long)b * (kL - 1) + (long long)(t0 - 1)) * 6;
  int lds_shift = 0;
  int count = kCH * 6;
  if (k == 0) { gstart += 6; lds_shift = 6; count -= 6; }  // avoid OOB at t=-1
  const float* gptr = eps_walk + gstart;                   // 24B-multiple => 8B aligned
  int npairs = count >> 1;                                 // count is even
  for (int i = (int)threadIdx.x; i < npairs; i += kThreads) {
    unsigned lds_addr =
        (unsigned)(uintptr_t)(lds_f*)(nlds + lds_shift + 2 * i);  // LDS byte offset
    unsigned voff = (unsigned)(8u * (unsigned)i);                 // global byte offset
    // GVS: global addr = gptr + voff; LDS dest = lds_addr   (ASYNCcnt)
    asm volatile("global_load_async_to_lds_b64 %0, %1, %2"
                 :: "v"(lds_addr), "v"(voff), "s"(gptr) : "memory");
  }
  asm volatile("s_wait_asynccnt 0" ::: "memory");
  __syncthreads();
  if (k == 0 && threadIdx.x < 6)
    nlds[threadIdx.x] = eps0[b * 6 + threadIdx.x] * (1.0f / DIFFS);
  __syncthreads();
}

// ---------------------------------------------------------------------------
// Pass 1: per-(batch,chunk) weighted reduction  S = sum_i a^{kCH-1-i} ds*n[i]
// ---------------------------------------------------------------------------
__global__ void __launch_bounds__(kThreads)
pass1_reduce(const float* __restrict__ eps_walk, const float* __restrict__ eps0,
             float* __restrict__ S) {
  __shared__ float nlds[kCH * 6];
  __shared__ float red[6 * 32];

  int blk = blockIdx.x;                // b * kNCH + k
  int b   = blk / kNCH;
  int k   = blk - b * kNCH;
  stage_n_async(nlds, eps_walk, eps0, b, k * kCH, k);

  int tid = threadIdx.x;
  if (tid < 192) {
    int c    = tid >> 5;               // channel 0..5
    int lane = tid & 31;               // residue class along t
    float A2  = A_COEF * A_COEF;
    float A4  = A2 * A2, A8 = A4 * A4, A16 = A8 * A8, A32 = A16 * A16;
    // fold a^{32}-weighted over elements i = lane + 32j (ascending j)
    float r = 0.0f;
    for (int j = 0; j < kCH / 32; ++j)
      r = fmaf(A32, r, nlds[(lane + 32 * j) * 6 + c]);
    // desired weight a^{2047-i} => extra a^{31-lane}; fold channel scale ds_c.
    r *= powi_a(31 - lane) * (std0_of(c) * DIFFS);
    red[c * 32 + lane] = r;
  }
  __syncthreads();
  if (tid < 6) {
    float s = 0.0f;
    for (int l = 0; l < 32; ++l) s += red[tid * 32 + l];
    S[(size_t)blk * 6 + tid] = s;
  }
}

// ---------------------------------------------------------------------------
// Pass 2: tiny serial scan of chunk sums -> per-chunk carry-in.
//   Cin_k = carry_{k-1},  carry_k = a^kCH * carry_{k-1} + S_k,  carry_{-1}=0
// ---------------------------------------------------------------------------
__global__ void pass2_carry(const float* __restrict__ S, float* __restrict__ Cin) {
  int tid = threadIdx.x;
  if (tid >= kB * 6) return;
  int b = tid / 6, c = tid - 6 * b;
  float ACH = A_COEF;                  // a^{2048} = a^(2^11): 11 squarings
  for (int i = 0; i < 11; ++i) ACH *= ACH;
  float carry = 0.0f;
  for (int k = 0; k < kNCH; ++k) {
    size_t off = ((size_t)b * kNCH + k) * 6 + c;
    Cin[off] = carry;
    carry = fmaf(ACH, carry, S[off]);
  }
}

// ---------------------------------------------------------------------------
// Pass 3: WMMA intra-chunk scan + elementwise combine.
//   Tile: D = (ds*P)(16x16, P[i][j]=a^{i-j} lower-tri) x N(16x16, col s = seg s)
//   via 4 chained V_WMMA_F32_16X16X4_F32 (K split 4x4).
//   bias[16s+i] = a^{i+1} H_s + D[i][s],  H_0 = chunk carry,
//   H_{s+1} = a^16 H_s + D[15][s].
// ---------------------------------------------------------------------------
__global__ void __launch_bounds__(kThreads)
pass3_apply(const float* __restrict__ imu, const float* __restrict__ eps_walk,
            const float* __restrict__ eps0, const float* __restrict__ meas,
            const float* __restrict__ Cin, float* __restrict__ out) {
  __shared__ float nlds[kCH * 6];      // raw increments; later reused for bias
  __shared__ float sums[6 * kSegs];    // per-segment weighted sums D[15][s]
  __shared__ float Hs[6 * kSegs];      // per-segment carries

  int blk = blockIdx.x;
  int b   = blk / kNCH;
  int k   = blk - b * kNCH;
  int t0  = k * kCH;
  stage_n_async(nlds, eps_walk, eps0, b, t0, k);

  int tid  = threadIdx.x;
  int wave = tid >> 5;                 // 0..7 (wave32)
  int lane = tid & 31;
  int col  = lane & 15;                // N-index (segment within tile) / M row
  int h    = lane >> 4;                // half-wave selector

  // --- A operand (constant matrix P) per ISA 32-bit 16x4 A layout:
  // lanes 0-15: VGPR0=K0, VGPR1=K1; lanes 16-31: VGPR0=K2, VGPR1=K3.
  // K-slice q covers columns j = 4q+2h, 4q+2h+1;  P[m][j] = a^{m-j} (m>=j).
  float invA = 1.0f / A_COEF;
  float acol = powi_a(col);            // a^{m}, m = col
  float ish  = h ? (invA * invA) : 1.0f;
  v2f Aop[4];
  float q4 = 1.0f;                     // invA^{4q}
#pragma unroll
  for (int q = 0; q < 4; ++q) {
    int e0 = col - 4 * q - 2 * h;
    float base = acol * q4 * ish;      // ~ a^{e0}
    Aop[q].x = (e0 >= 0) ? base : 0.0f;
    Aop[q].y = (e0 >= 1) ? base * invA : 0.0f;
    q4 *= invA * invA * invA * invA;
  }

  // --- Phase A: 48 tiles (6 ch x 8 tile-cols); wave w does tiles w, w+8, ...
  v8f D[6];
#pragma unroll
  for (int j = 0; j < 6; ++j) {
    int tile = wave + 8 * j;
    int c    = tile >> 3;              // channel
    int tcol = tile & 7;               // tile column: segments [16*tcol, +16)
    int tb   = tcol * 256;             // t_local base of tile
    float dsc = (c < 3 ? 0.2f : 0.015f) * DIFFS;  // fold OU diffusion scale into A
    v8f acc = {};
#pragma unroll
    for (int q = 0; q < 4; ++q) {
      // B layout (4x16): lanes 0-15 hold K=0,1 (VGPR0,1); lanes 16-31 K=2,3.
      int trow = tb + col * 16 + 4 * q + 2 * h;
      v2f Bop;
      Bop.x = nlds[trow * 6 + c];
      Bop.y = nlds[(trow + 1) * 6 + c];
      v2f As = Aop[q] * dsc;
      acc = __builtin_amdgcn_wmma_f32_16x16x4_f32(
          false, As, false, Bop, (short)0, acc, false, false);
    }
    D[j] = acc;
    // Row M=15 (segment sum) lives in VGPR7 of lanes 16-31.
    if (h) sums[c * kSegs + tcol * 16 + col] = acc[7];
  }
  __syncthreads();

  // --- Phase B: per-channel serial scan of 128 segment sums (cheap).
  if (tid < 6) {
    int c = tid;
    float A2 = A_COEF * A_COEF, A4 = A2 * A2, A8 = A4 * A4, A16 = A8 * A8;
    float hcar = Cin[((size_t)b * kNCH + k) * 6 + c];
    for (int s = 0; s < kSegs; ++s) {
      Hs[c * kSegs + s] = hcar;
      hcar = fmaf(A16, hcar, sums[c * kSegs + s]);
    }
  }
  __syncthreads();

  // --- Phase C: bias = a^{i+1} H_s + D[i][s]; store to LDS as [t][c].
  {
    float A2 = A_COEF * A_COEF, A4 = A2 * A2, A8 = A4 * A4;
#pragma unroll
    for (int j = 0; j < 6; ++j) {
      int tile = wave + 8 * j;
      int c    = tile >> 3;
      int tcol = tile & 7;
      int s    = tcol * 16 + col;
      float hval = Hs[c * kSegs + s];
      float ar = A_COEF * (h ? A8 : 1.0f);   // a^{i+1}, i = r + 8h
      int tbase = tcol * 256 + col * 16 + 8 * h;
#pragma unroll
      for (int r = 0; r < 8; ++r) {
        nlds[(tbase + r) * 6 + c] = fmaf(ar, hval, D[j][r]);
        ar *= A_COEF;
      }
    }
  }
  __syncthreads();

  // --- Phase D: coalesced combine & write-out.
  const float invL1   = 1.0f / 65535.0f;     // linspace(0,1,L): t/(L-1)
  const float TWO_PI  = 6.283185307179586f;
  for (int tl = tid; tl < kCH; tl += kThreads) {
    int t = t0 + tl;
    float tt   = (float)t * invL1;
    float temp = 20.0f + 5.0f * sinf(TWO_PI * tt) + 2.0f * tt;
    size_t base = ((size_t)b * kL + (size_t)t) * 6;
#pragma unroll
    for (int c = 0; c < 6; ++c) {
      float ns = (c < 3) ? 0.1f   : 0.01f;   // measurement noise std
      float tc = (c < 3) ? 0.001f : 0.01f;   // 0.1*TEMP_COEF / TEMP_COEF
      out[base + c] = imu[base + c] + nlds[tl * 6 + c] +
                      meas[base + c] * ns + temp * tc;
    }
  }
}

}  // namespace

extern "C" void kernel_launch(void* const* d_in, const int* in_sizes, int n_in,
                              void* d_out, int out_size, void* d_ws, size_t ws_size,
                              hipStream_t stream) {
  const float* imu      = (const float*)d_in[0];  // [B, T, K, 6]
  const float* eps0     = (const float*)d_in[1];  // [B, 6]
  const float* eps_walk = (const float*)d_in[2];  // [B, L-1, 6]
  const float* meas     = (const float*)d_in[3];  // [B, L, 6]
  float* out = (float*)d_out;

  float* S   = (float*)d_ws;                      // [B*kNCH*6]
  float* Cin = S + (size_t)kB * kNCH * 6;         // [B*kNCH*6]

  dim3 grid(kB * kNCH);
  pass1_reduce<<<grid, kThreads, 0, stream>>>(eps_walk, eps0, S);
  pass2_carry<<<1, 192, 0, stream>>>(S, Cin);
  pass3_apply<<<grid, kThreads, 0, stream>>>(imu, eps_walk, eps0, meas, Cin, out);
}